// HyperbolicGraphConvolution_24962349924474
// MI455X (gfx1250) — compile-verified
//
#include <hip/hip_runtime.h>
#include <hip/hip_bf16.h>
#include <stdint.h>

#define DIM   64
#define EB    8     // edges per async batch (one async b64 gather instr per edge)
#define CHUNK 128   // edges per wave (NNZ = 6.4M divides exactly)
#define WPB   8     // waves per 256-thread block (wave32)

typedef float v2f __attribute__((ext_vector_type(2)));
typedef float v8f __attribute__((ext_vector_type(8)));

// ---------------- logmap0 on the hyperboloid (CURVE = 1) ----------------
// One wave32 per 16 rows. Row sum-of-squares computed on the matrix pipe in
// full f32: 16 chained V_WMMA_F32_16X16X4_F32 accumulate X*X^T (Gram) over the
// 64-dim K; the diagonal is the exact per-row ||x||^2. The A-register layout
// (lane l<16: row l, K{0,1}; lane l+16: row l, K{2,3}) doubles as the B layout
// of X^T, so A and B are literally the same VGPRs.
__global__ __launch_bounds__(128) void logmap_kernel(const float* __restrict__ x,
                                                     float* __restrict__ h, int n) {
  const int wave = (int)((blockIdx.x * blockDim.x + threadIdx.x) >> 5);
  const int lane = threadIdx.x & 31;
  const int row0 = wave * 16;
  if (row0 >= n) return;

  if (row0 + 16 <= n) {
    const int r    = lane & 15;   // row within the 16-row group
    const int half = lane >> 4;   // which K-pair of the 4-wide chunk
    const int row  = row0 + r;
    v2f av[16];
    v8f c = {};
#pragma unroll
    for (int q = 0; q < 16; ++q) {
      av[q] = *(const v2f*)(x + (size_t)row * DIM + 4 * q + 2 * half);
      // D = A * A^T + C  (A == B registers), f32 exact
      c = __builtin_amdgcn_wmma_f32_16x16x4_f32(false, av[q], false, av[q],
                                                (short)0, c, false, false);
    }
    // diag(X*X^T): row r lives at vgpr (r&7), lane (r<8 ? r : r+16).
    // Each lane first selects the component its *own* row indexes; the source
    // lane's selection is exactly the diagonal entry we need, then one shuffle.
    float comp = c[0];
    if ((r & 7) == 1) comp = c[1];
    if ((r & 7) == 2) comp = c[2];
    if ((r & 7) == 3) comp = c[3];
    if ((r & 7) == 4) comp = c[4];
    if ((r & 7) == 5) comp = c[5];
    if ((r & 7) == 6) comp = c[6];
    if ((r & 7) == 7) comp = c[7];
    const int srcLane = (r < 8) ? r : (r + 16);
    float ss = __shfl(comp, srcLane);          // sum of squares over all 64 dims
    float x0 = __shfl(av[0].x, r);             // x[row][0] (lane r<16, chunk 0 .x)
    float yn2   = fmaxf(ss - x0 * x0, 0.0f);   // ||y||^2 (exclude dim 0)
    float norm  = fmaxf(sqrtf(yn2), 1e-15f);
    float theta = fmaxf(x0, 1.0f + 1e-7f);
    float s     = acoshf(theta) / norm;
#pragma unroll
    for (int q = 0; q < 16; ++q) {
      v2f o;
      o.x = s * av[q].x;
      o.y = s * av[q].y;
      if (q == 0 && half == 0) o.x = 0.0f;     // h[:,0] = 0
      *(v2f*)(h + (size_t)row * DIM + 4 * q + 2 * half) = o;
    }
  } else {
    // scalar fallback: wave-per-row shuffle reduction (n not multiple of 16)
    for (int rr = row0; rr < n; ++rr) {
      float2 v = ((const float2*)(x + (size_t)rr * DIM))[lane];
      float ssl = v.y * v.y + ((lane == 0) ? 0.0f : v.x * v.x);
#pragma unroll
      for (int m = 16; m >= 1; m >>= 1) ssl += __shfl_xor(ssl, m);
      const float x0l   = __shfl(v.x, 0);
      const float norm  = fmaxf(sqrtf(ssl), 1e-15f);
      const float theta = fmaxf(x0l, 1.0f + 1e-7f);
      const float s     = acoshf(theta) / norm;
      float2 o;
      o.x = (lane == 0) ? 0.0f : s * v.x;
      o.y = s * v.y;
      ((float2*)(h + (size_t)rr * DIM))[lane] = o;
    }
  }
}

// ---------------- elementwise helpers (float4) ----------------
__global__ __launch_bounds__(256) void zero_kernel(float4* __restrict__ p, int n4) {
  int i = blockIdx.x * blockDim.x + threadIdx.x;
  if (i < n4) { float4 z; z.x = z.y = z.z = z.w = 0.0f; p[i] = z; }
}
__global__ __launch_bounds__(256) void copy_kernel(float4* __restrict__ d,
                                                   const float4* __restrict__ s, int n4) {
  int i = blockIdx.x * blockDim.x + threadIdx.x;
  if (i < n4) d[i] = s[i];
}
// t = a + b; b = t; o = t   (computes S, seeds next scatter target, keeps S)
__global__ __launch_bounds__(256) void sum_dup_kernel(float4* __restrict__ b,
                                                      const float4* __restrict__ a,
                                                      float4* __restrict__ o, int n4) {
  int i = blockIdx.x * blockDim.x + threadIdx.x;
  if (i < n4) {
    float4 va = a[i], vb = b[i], t;
    t.x = va.x + vb.x; t.y = va.y + vb.y; t.z = va.z + vb.z; t.w = va.w + vb.w;
    b[i] = t; o[i] = t;
  }
}
__global__ __launch_bounds__(256) void add_kernel(float4* __restrict__ o,
                                                  const float4* __restrict__ b, int n4) {
  int i = blockIdx.x * blockDim.x + threadIdx.x;
  if (i < n4) {
    float4 vo = o[i], vb = b[i];
    vo.x += vb.x; vo.y += vb.y; vo.z += vb.z; vo.w += vb.w;
    o[i] = vo;
  }
}

// low 32 bits of a flat pointer into the LDS aperture == LDS byte offset
__device__ __forceinline__ uint32_t lds_lo32(const void* p) {
  return (uint32_t)(uintptr_t)p;
}

// Issue EB async global->LDS gathers: one instruction per edge, each gathers a
// full 256B source row (per-lane b64) into LDS. Tracked on ASYNCcnt.
__device__ __forceinline__ void issue_gather(const int* __restrict__ cols,
                                             long ebase, const float* __restrict__ prev,
                                             int lane, uint32_t lds_base) {
#pragma unroll
  for (int i = 0; i < EB; ++i) {
    int c = cols[ebase + i];                                  // wave-uniform -> s_load
    const float* gp = prev + (size_t)c * DIM + (lane << 1);
    uint32_t dst = lds_base + (uint32_t)(i * DIM * 4) + (uint32_t)(lane << 3);
    asm volatile("global_load_async_to_lds_b64 %0, %1, off"
                 :: "v"(dst), "v"(gp) : "memory");
  }
}

// cur[rows[e]][:] += vals[e] * prev[cols[e]][:]  — one wave per 128-edge chunk,
// double-buffered async LDS gather of source rows, coalesced 256B atomic bursts.
// prev/cur (51.2 MB each) are L2-resident on MI455X (192 MB L2).
__global__ __launch_bounds__(256) void scatter_kernel(
    const int* __restrict__ rows, const int* __restrict__ cols,
    const float* __restrict__ vals, const float* __restrict__ prev,
    float* __restrict__ cur, int nnz) {
  __shared__ float stage[WPB * 2 * EB * DIM];
  const int wid  = threadIdx.x >> 5;
  const int lane = threadIdx.x & 31;
  const long e0  = ((long)blockIdx.x * WPB + wid) * CHUNK;
  if (e0 >= nnz) return;

  float* buf0 = stage + wid * (2 * EB * DIM);
  float* buf1 = buf0 + EB * DIM;

  if (e0 + CHUNK <= nnz) {
    const uint32_t b0 = lds_lo32(buf0);
    const uint32_t b1 = lds_lo32(buf1);
    const int NB = CHUNK / EB;                 // 16 full batches
    issue_gather(cols, e0, prev, lane, b0);    // prime pipeline
    for (int b = 0; b < NB; ++b) {
      if (b + 1 < NB) {
        issue_gather(cols, e0 + (long)(b + 1) * EB, prev, lane,
                     ((b + 1) & 1) ? b1 : b0);
        asm volatile("s_wait_asynccnt 8" ::: "memory");   // batch b done, b+1 in flight
      } else {
        asm volatile("s_wait_asynccnt 0" ::: "memory");   // drain
      }
      const float* buf = (b & 1) ? buf1 : buf0;
      const long ebase = e0 + (long)b * EB;
#pragma unroll
      for (int i = 0; i < EB; ++i) {
        int   r = rows[ebase + i];
        float w = vals[ebase + i];
        float2 g = ((const float2*)(buf + i * DIM))[lane];
        float* dp = cur + (size_t)r * DIM + (lane << 1);
        atomicAdd(dp,     w * g.x);
        atomicAdd(dp + 1, w * g.y);
      }
    }
  } else {
    // generic tail (not hit for NNZ = 6.4M, kept for safety)
    for (long e = e0; e < nnz; ++e) {
      int c = cols[e]; int r = rows[e]; float w = vals[e];
      float2 g = ((const float2*)(prev + (size_t)c * DIM))[lane];
      float* dp = cur + (size_t)r * DIM + (lane << 1);
      atomicAdd(dp,     w * g.x);
      atomicAdd(dp + 1, w * g.y);
    }
  }
}

extern "C" void kernel_launch(void* const* d_in, const int* in_sizes, int n_in,
                              void* d_out, int out_size, void* d_ws, size_t ws_size,
                              hipStream_t stream) {
  const float* x    = (const float*)d_in[0];
  const int*   rows = (const int*)d_in[1];
  const int*   cols = (const int*)d_in[2];
  const float* vals = (const float*)d_in[3];
  const int nnz = in_sizes[1];
  const int n   = in_sizes[0] / DIM;

  float* out = (float*)d_out;
  float* A   = (float*)d_ws;                 // n*DIM floats
  float* B   = A + (size_t)n * DIM;          // n*DIM floats

  const int n4    = (n * DIM) / 4;
  dim3 blk(256);
  const int egrid = (n4 + 255) / 256;
  const int lgrid = ((n + 15) / 16 + 3) / 4;  // 16 rows/wave, 4 waves/block (128 thr)
  const int sgrid = (int)(((long)nnz + (long)CHUNK * WPB - 1) / ((long)CHUNK * WPB));

  // prev0 = h = logmap0(x)   (WMMA Gram-diagonal row norms)
  logmap_kernel<<<lgrid, dim3(128), 0, stream>>>(x, A, n);
  // layer 0: B = spmm(A) = cur1 = S1 ; prev = B
  zero_kernel<<<egrid, blk, 0, stream>>>((float4*)B, n4);
  scatter_kernel<<<sgrid, blk, 0, stream>>>(rows, cols, vals, A, B, nnz);
  // layer 1: A := S1 ; A += spmm(B)  -> cur2 ; prev = A
  copy_kernel<<<egrid, blk, 0, stream>>>((float4*)A, (const float4*)B, n4);
  scatter_kernel<<<sgrid, blk, 0, stream>>>(rows, cols, vals, B, A, nnz);
  // S2 = S1 + cur2 = B + A ; keep in out, seed B := S2 for layer 2
  sum_dup_kernel<<<egrid, blk, 0, stream>>>((float4*)B, (const float4*)A, (float4*)out, n4);
  // layer 2: B += spmm(A) -> cur3 ; prev = B
  scatter_kernel<<<sgrid, blk, 0, stream>>>(rows, cols, vals, A, B, nnz);
  // S3 = S2 + cur3 = out + B ; seed out := S3 for final layer
  add_kernel<<<egrid, blk, 0, stream>>>((float4*)out, (const float4*)B, n4);
  // layer 3: out += spmm(B) -> cur4 = answer
  scatter_kernel<<<sgrid, blk, 0, stream>>>(rows, cols, vals, B, out, nnz);
}